// SparseMOE_72335839199354
// MI455X (gfx1250) — compile-verified
//
#include <hip/hip_runtime.h>
#include <hip/hip_bf16.h>
#include <math.h>

typedef __attribute__((ext_vector_type(2))) float v2f;
typedef __attribute__((ext_vector_type(8))) float v8f;

constexpr int kD = 1024;   // hidden dim
constexpr int kE = 16;     // routed experts
constexpr int kN = 8192;   // B*T tokens
constexpr float kLambda = 0.1f;

// ---------------- kernel 1: zero h2 scratch ----------------
__global__ __launch_bounds__(1024) void zero_h2_kernel(float* __restrict__ h2) {
    h2[threadIdx.x] = 0.0f;
}

// ---------------- kernel 2: h2[d] = sum_n x[n,d]^2 ----------------
// grid (kD/256, 32), block 256. Coalesced column reads, fp32 global atomics.
__global__ __launch_bounds__(256) void h2_reduce_kernel(const float* __restrict__ x,
                                                        float* __restrict__ h2) {
    const int c  = blockIdx.x * 256 + threadIdx.x;
    const int r0 = blockIdx.y * 256;
    const float* p = x + (size_t)r0 * kD + c;
    float s = 0.0f;
    #pragma unroll 8
    for (int r = 0; r < 256; ++r) {
        float v = p[(size_t)r * kD];
        s = fmaf(v, v, s);
    }
    atomicAdd(&h2[c], s);
}

// ---------------- kernel 3: Gram matrix via WMMA + diversity loss ----------------
// One wave32. G[e,f] = sum_d h2[d]*re[e,d]*re[f,d], accumulated with
// V_WMMA_F32_16X16X4_F32 (A = re*h2, B = re^T). G is symmetric -> layout-robust.
__global__ __launch_bounds__(32) void gram_div_kernel(const float* __restrict__ re,
                                                      const float* __restrict__ h2,
                                                      float* __restrict__ out_loss) {
    __shared__ float G[kE][kE];
    const int lane = threadIdx.x;
    const int m    = lane & 15;            // A-row / B-col index
    const int ko   = (lane >> 4) << 1;     // 0 for lanes 0-15, 2 for lanes 16-31
    const float* rr = re + (size_t)m * kD + ko;
    const float* hh = h2 + ko;

    v8f c = {};
    for (int kb = 0; kb < kD; kb += 4) {
        float r0 = rr[kb], r1 = rr[kb + 1];
        v2f a, b;
        b.x = r0;            b.y = r1;
        a.x = r0 * hh[kb];   a.y = r1 * hh[kb + 1];
        c = __builtin_amdgcn_wmma_f32_16x16x4_f32(false, a, false, b,
                                                  (short)0, c, false, false);
    }
    const int mrow = (lane >> 4) * 8;
    #pragma unroll
    for (int v = 0; v < 8; ++v) G[mrow + v][m] = c[v];
    __syncthreads();

    if (lane == 0) {
        float nrm[kE];
        #pragma unroll
        for (int e = 0; e < kE; ++e)
            nrm[e] = fmaxf(sqrtf(fmaxf(G[e][e], 0.0f)), 1e-8f);
        float acc = 0.0f;
        for (int e = 0; e < kE; ++e)
            for (int f = 0; f < kE; ++f) {
                if (e == f) continue;
                float sim = G[e][f] / (nrm[e] * nrm[f]);
                acc += fminf(fmaxf(sim, -1.0f), 1.0f);
            }
        out_loss[0] = acc / (float)(kE * (kE - 1)) * kLambda;
    }
}

// ---------------- kernel 4: fused router (WMMA) + softmax/top2 + output ----------
// Block = 256 threads (8 waves), handles 16 tokens. Each wave computes the
// 16x16 logits tile over a K=128 slice of D with 32 fp32 WMMAs; partial C
// tiles are combined via LDS ds_add_f32 atomics. Then per-token softmax/top-2
// and a float4-vectorized output combine (one float4 per thread per token).
__global__ __launch_bounds__(256) void moe_fused_kernel(const float* __restrict__ x,
                                                        const float* __restrict__ gate_w,
                                                        const float* __restrict__ gate_b,
                                                        const float* __restrict__ se,
                                                        const float* __restrict__ re,
                                                        float* __restrict__ out) {
    __shared__ float slog[kE * kE];      // [token][expert]
    __shared__ float sw0[16], sw1[16];
    __shared__ int   si0[16], si1[16];

    const int tid = threadIdx.x;
    const int t0  = blockIdx.x * 16;

    if (tid < kE * kE) slog[tid] = 0.0f;
    __syncthreads();

    // ---- phase 1: logits tile via WMMA f32 16x16x4 ----
    const int lane = tid & 31;
    const int wv   = tid >> 5;            // wave id: K slice [wv*128, wv*128+128)
    const int m    = lane & 15;
    const int ko   = (lane >> 4) << 1;
    const float* xrow = x      + (size_t)(t0 + m) * kD + (wv * 128 + ko);
    const float* grow = gate_w + (size_t)m * kD        + (wv * 128 + ko);

    v8f c = {};
    #pragma unroll 4
    for (int it = 0; it < 32; ++it) {
        const int kb = it * 4;
        v2f a, b;
        a.x = xrow[kb]; a.y = xrow[kb + 1];
        b.x = grow[kb]; b.y = grow[kb + 1];
        c = __builtin_amdgcn_wmma_f32_16x16x4_f32(false, a, false, b,
                                                  (short)0, c, false, false);
    }
    const int mrow = (lane >> 4) * 8;
    #pragma unroll
    for (int v = 0; v < 8; ++v)
        atomicAdd(&slog[(mrow + v) * kE + m], c[v]);
    __syncthreads();

    // ---- phase 2: per-token softmax + top-2 weights ----
    if (tid < 16) {
        float l[kE];
        float mx = -3.4e38f;
        #pragma unroll
        for (int e = 0; e < kE; ++e) {
            l[e] = slog[tid * kE + e] + gate_b[e];
            mx = fmaxf(mx, l[e]);
        }
        float sum = 0.0f;
        #pragma unroll
        for (int e = 0; e < kE; ++e) { l[e] = expf(l[e] - mx); sum += l[e]; }
        float p0 = -1.0f, p1 = -1.0f; int i0 = 0, i1 = 0;
        #pragma unroll
        for (int e = 0; e < kE; ++e) {
            float p = l[e];
            if (p > p0)      { p1 = p0; i1 = i0; p0 = p; i0 = e; }
            else if (p > p1) { p1 = p;  i1 = e; }
        }
        const float inv = 1.0f / sum;
        const float v0 = p0 * inv, v1 = p1 * inv;
        const float s  = fmaxf(v0 + v1, 1e-6f);
        sw0[tid] = v0 / s; sw1[tid] = v1 / s;
        si0[tid] = i0;     si1[tid] = i1;
    }
    __syncthreads();

    // ---- phase 3: out = x * (sum(se) + w0*re[i0] + w1*re[i1]) ----
    const float4* x4  = (const float4*)x;
    const float4* se4 = (const float4*)se;
    const float4* re4 = (const float4*)re;
    float4* out4 = (float4*)out;

    const float4 sA = se4[tid];
    const float4 sB = se4[(kD / 4) + tid];
    const float4 sw = make_float4(sA.x + sB.x, sA.y + sB.y, sA.z + sB.z, sA.w + sB.w);

    for (int r = 0; r < 16; ++r) {
        const float w0 = sw0[r], w1 = sw1[r];
        const int   i0 = si0[r], i1 = si1[r];
        const size_t row = (size_t)(t0 + r) * (kD / 4) + tid;
        const float4 xr = x4[row];
        const float4 e0 = re4[(size_t)i0 * (kD / 4) + tid];
        const float4 e1 = re4[(size_t)i1 * (kD / 4) + tid];
        float4 o;
        o.x = xr.x * (sw.x + w0 * e0.x + w1 * e1.x);
        o.y = xr.y * (sw.y + w0 * e0.y + w1 * e1.y);
        o.z = xr.z * (sw.z + w0 * e0.z + w1 * e1.z);
        o.w = xr.w * (sw.w + w0 * e0.w + w1 * e1.w);
        out4[row] = o;
    }
}

// ---------------- launcher ----------------
extern "C" void kernel_launch(void* const* d_in, const int* in_sizes, int n_in,
                              void* d_out, int out_size, void* d_ws, size_t ws_size,
                              hipStream_t stream) {
    const float* x      = (const float*)d_in[0];   // [4,2048,1024]
    const float* gate_w = (const float*)d_in[1];   // [16,1024]
    const float* gate_b = (const float*)d_in[2];   // [16]
    const float* se     = (const float*)d_in[3];   // [2,1024]
    const float* re     = (const float*)d_in[4];   // [16,1024]
    float* out = (float*)d_out;                    // [N*D] output + [1] div_loss
    float* h2  = (float*)d_ws;                     // kD floats of scratch

    zero_h2_kernel<<<1, 1024, 0, stream>>>(h2);
    h2_reduce_kernel<<<dim3(kD / 256, kN / 256), 256, 0, stream>>>(x, h2);
    gram_div_kernel<<<1, 32, 0, stream>>>(re, h2, out + (size_t)kN * kD);
    moe_fused_kernel<<<kN / 16, 256, 0, stream>>>(x, gate_w, gate_b, se, re, out);
}